// FIStateProbabilitiesPaulied_22024592293963
// MI455X (gfx1250) — compile-verified
//
#include <hip/hip_runtime.h>
#include <hip/hip_bf16.h>

typedef __attribute__((ext_vector_type(2))) float v2f;
typedef __attribute__((ext_vector_type(4))) float v4f;
typedef __attribute__((ext_vector_type(8))) float v8f;

#define NDRIVES 4
#define LPAULI  64
#define DIM     8
#define NTHREADS 128

__global__ __launch_bounds__(NTHREADS)
void fi_fisher_kernel(const double* __restrict__ x,
                      const double* __restrict__ kern,
                      const double* __restrict__ bias,
                      const double* __restrict__ paulies,  // [64][8][8] complex128 = [64][64][2] double
                      float* __restrict__ out, int batch)
{
    const int bid = blockIdx.x;
    const int tid = threadIdx.x;

    __shared__ double pw[LPAULI];
    __shared__ double xs[NDRIVES];
    __shared__ double Hre[DIM][DIM], Him[DIM][DIM];
    __shared__ double Vre[DIM][DIM], Vim[DIM][DIM];
    __shared__ double ev[DIM];
    __shared__ double rc[4], rwre[4], rwim[4];
    __shared__ int    rp[4], rq[4];
    __shared__ double ampre[DIM], ampim[DIM], Pv[DIM];
    __shared__ double eer[DIM], eei[DIM];            // exp(-i e)
    __shared__ double Gmr[DIM][DIM], Gmi[DIM][DIM];  // Gamma (divided differences)
    __shared__ double Zr[DIM][DIM],  Zi[DIM][DIM];   // Z[s][b] = conj(V[0][b]) V[s][b]
    __shared__ double M2r[DIM][DIM], M2i[DIM][DIM];  // M2[a][s] = sum_b Gamma[a][b] Z[s][b]
    __shared__ double Wr[DIM][DIM][DIM], Wi[DIM][DIM][DIM];
    __shared__ float  Gsf[DIM][LPAULI];              // g[p][k]/sqrt(P_p) as f32
    __shared__ float  Ib[LPAULI][LPAULI];            // I_b for this sample

    // warm L2 with paulies (gfx1250 global_prefetch path)
    __builtin_prefetch(paulies + (size_t)tid * 64, 0, 1);

    if (tid < NDRIVES) xs[tid] = x[(size_t)bid * NDRIVES + tid];
    __syncthreads();

    // pw = x @ kernel + bias
    if (tid < LPAULI) {
        double s = bias[tid];
        for (int j = 0; j < NDRIVES; ++j) s += xs[j] * kern[j * LPAULI + tid];
        pw[tid] = s;
    }
    __syncthreads();

    // H = sum_k pw[k] * Pauli_k ; V = I
    if (tid < 64) {
        int r = tid >> 3, s = tid & 7;
        double hr = 0.0, hi = 0.0;
        for (int k = 0; k < LPAULI; ++k) {
            const double* pk = paulies + ((size_t)k * 64 + tid) * 2;
            hr += pw[k] * pk[0];
            hi += pw[k] * pk[1];
        }
        Hre[r][s] = hr; Him[r][s] = hi;
        Vre[r][s] = (r == s) ? 1.0 : 0.0;
        Vim[r][s] = 0.0;
    }
    __syncthreads();

    // ---- complex Jacobi eigensolve, tournament schedule: 7 rounds x 4 disjoint pairs ----
    int arr[8] = {0, 1, 2, 3, 4, 5, 6, 7};
    for (int sweep = 0; sweep < 8; ++sweep) {
        for (int round = 0; round < 7; ++round) {
            if (tid < 4) {
                int p = arr[tid], q = arr[7 - tid];
                if (p > q) { int t = p; p = q; q = t; }
                double ar = Hre[p][q], ai = Him[p][q];
                double mag = sqrt(ar * ar + ai * ai);
                double c = 1.0, wre = 0.0, wim = 0.0;
                if (mag > 1e-13) {
                    double phr = ar / mag, phi = ai / mag;
                    double tau = (Hre[q][q] - Hre[p][p]) / (2.0 * mag);
                    double tt  = ((tau >= 0.0) ? 1.0 : -1.0) / (fabs(tau) + sqrt(1.0 + tau * tau));
                    c = 1.0 / sqrt(1.0 + tt * tt);
                    double s_ = tt * c;
                    wre = s_ * phr; wim = s_ * phi;
                }
                rc[tid] = c; rwre[tid] = wre; rwim[tid] = wim; rp[tid] = p; rq[tid] = q;
            }
            __syncthreads();
            // right-multiply: H <- H*J, V <- V*J  (one thread per row)
            if (tid < DIM) {
                int r = tid;
                for (int i = 0; i < 4; ++i) {
                    int p = rp[i], q = rq[i];
                    double c = rc[i], wre = rwre[i], wim = rwim[i];
                    double hpr = Hre[r][p], hpi = Him[r][p], hqr = Hre[r][q], hqi = Him[r][q];
                    Hre[r][p] = c * hpr - (wre * hqr + wim * hqi);
                    Him[r][p] = c * hpi - (wre * hqi - wim * hqr);
                    Hre[r][q] = (wre * hpr - wim * hpi) + c * hqr;
                    Him[r][q] = (wre * hpi + wim * hpr) + c * hqi;
                    double vpr = Vre[r][p], vpi = Vim[r][p], vqr = Vre[r][q], vqi = Vim[r][q];
                    Vre[r][p] = c * vpr - (wre * vqr + wim * vqi);
                    Vim[r][p] = c * vpi - (wre * vqi - wim * vqr);
                    Vre[r][q] = (wre * vpr - wim * vpi) + c * vqr;
                    Vim[r][q] = (wre * vpi + wim * vpr) + c * vqi;
                }
            }
            __syncthreads();
            // left-multiply: H <- J^H * H  (one thread per column)
            if (tid < DIM) {
                int cc = tid;
                for (int i = 0; i < 4; ++i) {
                    int p = rp[i], q = rq[i];
                    double c = rc[i], wre = rwre[i], wim = rwim[i];
                    double mpr = Hre[p][cc], mpi = Him[p][cc], mqr = Hre[q][cc], mqi = Him[q][cc];
                    Hre[p][cc] = c * mpr - (wre * mqr - wim * mqi);
                    Him[p][cc] = c * mpi - (wre * mqi + wim * mqr);
                    Hre[q][cc] = (wre * mpr + wim * mpi) + c * mqr;
                    Him[q][cc] = (wre * mpi - wim * mpr) + c * mqi;
                }
            }
            __syncthreads();
            int last = arr[7];
            for (int i = 7; i > 1; --i) arr[i] = arr[i - 1];
            arr[1] = last;
        }
    }

    // eigenvalues, exp(-i e), amplitudes, probabilities
    if (tid < DIM) {
        double e = Hre[tid][tid];
        ev[tid]  = e;
        eer[tid] = cos(e);
        eei[tid] = -sin(e);
    }
    __syncthreads();
    if (tid < DIM) {
        int p = tid;
        double sr = 0.0, si = 0.0;
        for (int s = 0; s < DIM; ++s) {
            double vr = Vre[p][s], vi = Vim[p][s];
            double ur = Vre[0][s], ui = -Vim[0][s];   // conj(V[0][s])
            double tr = vr * ur - vi * ui, ti = vr * ui + vi * ur;
            sr += tr * eer[s] - ti * eei[s];
            si += tr * eei[s] + ti * eer[s];
        }
        ampre[p] = sr; ampim[p] = si; Pv[p] = sr * sr + si * si;
    }
    __syncthreads();

    // Gamma (Daleckii-Krein divided differences for f(x)=exp(-ix)) and Z
    if (tid < 64) {
        int a = tid >> 3, b = tid & 7;
        double d = ev[a] - ev[b];
        double gr, gi;
        if (fabs(d) > 1e-9) {
            gr = (eer[a] - eer[b]) / d;
            gi = (eei[a] - eei[b]) / d;
        } else {            // f'(e_a) = -i * exp(-i e_a)
            gr = eei[a];
            gi = -eer[a];
        }
        Gmr[a][b] = gr; Gmi[a][b] = gi;
        int s = a, bb = b;
        double ur = Vre[0][bb], ui = -Vim[0][bb];
        double vr = Vre[s][bb], vi = Vim[s][bb];
        Zr[s][bb] = ur * vr - ui * vi;
        Zi[s][bb] = ur * vi + ui * vr;
    }
    __syncthreads();
    if (tid < 64) {
        int a = tid >> 3, s = tid & 7;
        double mr = 0.0, mi = 0.0;
        for (int b = 0; b < DIM; ++b) {
            double gr = Gmr[a][b], gi = Gmi[a][b], zr = Zr[s][b], zi = Zi[s][b];
            mr += gr * zr - gi * zi;
            mi += gr * zi + gi * zr;
        }
        M2r[a][s] = mr; M2i[a][s] = mi;
    }
    __syncthreads();

    // W^(p)[r][s] = sum_a conj(V[r][a]) * (conj(amp_p) V[p][a]) * M2[a][s]
    for (int it = 0; it < 4; ++it) {
        int idx = tid + NTHREADS * it;          // 0..511
        int p = idx >> 6, r = (idx >> 3) & 7, s = idx & 7;
        double car = ampre[p], cai = -ampim[p];
        double wr_ = 0.0, wi_ = 0.0;
        for (int a = 0; a < DIM; ++a) {
            double vr = Vre[p][a], vi = Vim[p][a];
            double yr = car * vr - cai * vi, yi = car * vi + cai * vr;
            double cvr = Vre[r][a], cvi = -Vim[r][a];
            double tr = cvr * yr - cvi * yi, ti = cvr * yi + cvi * yr;
            double mr = M2r[a][s], mi = M2i[a][s];
            wr_ += tr * mr - ti * mi;
            wi_ += tr * mi + ti * mr;
        }
        Wr[p][r][s] = wr_; Wi[p][r][s] = wi_;
    }
    __syncthreads();

    // g[p][k] = 2*Re( sum_{rs} S_k[r][s] * W^(p)[r][s] ) ; scale by 1/sqrt(P_p)
    for (int it = 0; it < 4; ++it) {
        int idx = tid + NTHREADS * it;          // (p,k)
        int p = idx >> 6, k = idx & 63;
        const double* S = paulies + (size_t)k * 128;
        double acc = 0.0;
        for (int e = 0; e < 64; ++e) {
            int r = e >> 3, s = e & 7;
            acc += S[2 * e] * Wr[p][r][s] - S[2 * e + 1] * Wi[p][r][s];
        }
        Gsf[p][k] = (float)((2.0 * acc) * rsqrt(Pv[p]));
    }
    __syncthreads();

    // ---- I_b = Gs^T * Gs via V_WMMA_F32_16X16X4_F32 (wave w owns 16x64 strip) ----
    {
        int wv   = tid >> 5;     // 0..3 = k-tile
        int lane = tid & 31;
        int row  = lane & 15;
        int half = lane >> 4;
        v8f acc[4] = {};
        for (int kstep = 0; kstep < 2; ++kstep) {
            int p0 = kstep * 4 + half * 2;      // K-pair held by this half-wave
            v2f a;
            a.x = Gsf[p0][wv * 16 + row];
            a.y = Gsf[p0 + 1][wv * 16 + row];
            for (int mt = 0; mt < 4; ++mt) {
                v2f b;
                b.x = Gsf[p0][mt * 16 + row];
                b.y = Gsf[p0 + 1][mt * 16 + row];
                acc[mt] = __builtin_amdgcn_wmma_f32_16x16x4_f32(
                    false, a, false, b, (short)0, acc[mt], false, false);
            }
        }
        for (int mt = 0; mt < 4; ++mt)
            for (int r = 0; r < 8; ++r)
                Ib[wv * 16 + r + 8 * half][mt * 16 + row] = acc[mt][r];
    }
    __syncthreads();

    // ---- write outputs, bandwidth-optimized ----
    // I_k[i,j,k,l,m] = x[j]x[l]*I_b[k,m] (67 MB stream, NT B128 stores), plus I_b.
    // Each float4 of Ib is read from LDS once and fanned out to 17 stores.
    {
        size_t ikbase = (size_t)bid * 65536;
        size_t ibbase = (size_t)batch * 65536 + (size_t)bid * 4096;
        float xf[NDRIVES];
        for (int j = 0; j < NDRIVES; ++j) xf[j] = (float)xs[j];
        float sc[16];
        for (int jl = 0; jl < 16; ++jl) sc[jl] = xf[jl >> 2] * xf[jl & 3];

        for (int c = tid; c < 1024; c += NTHREADS) {   // 1024 float4 chunks of Ib
            int k  = c >> 4;
            int m4 = (c & 15) << 2;
            v4f val = *(const v4f*)&Ib[k][m4];
            // I_b (re-read later by nothing, but small: keep RT stores)
            *(v4f*)&out[ibbase + (size_t)k * 64 + m4] = val;
            // 16 scaled copies into I_k, non-temporal (streamed once)
            size_t kmoff = (size_t)k * 256 + m4;
            for (int jl = 0; jl < 16; ++jl) {
                int j = jl >> 2, l = jl & 3;
                v4f sv = val * sc[jl];
                __builtin_nontemporal_store(
                    sv, (v4f*)&out[ikbase + (size_t)j * 16384 + (size_t)l * 64 + kmoff]);
            }
        }
    }
}

extern "C" void kernel_launch(void* const* d_in, const int* in_sizes, int n_in,
                              void* d_out, int out_size, void* d_ws, size_t ws_size,
                              hipStream_t stream) {
    const double* x       = (const double*)d_in[0];
    const double* kern    = (const double*)d_in[1];
    const double* bias    = (const double*)d_in[2];
    const double* paulies = (const double*)d_in[3];   // complex128 as interleaved doubles
    float* out = (float*)d_out;
    int batch = in_sizes[0] / NDRIVES;                // 256
    fi_fisher_kernel<<<batch, NTHREADS, 0, stream>>>(x, kern, bias, paulies, out, batch);
}